// GraphLayer_88596585382212
// MI455X (gfx1250) — compile-verified
//
#include <hip/hip_runtime.h>
#include <hip/hip_bf16.h>

// ---------------------------------------------------------------------------
// MI455X (gfx1250) implementation.
//
// Roofline reasoning: the naive W@G is 2*8192^2*256 = 34.4 GFLOP + a 256MB
// intermediate. leaky(s_i + t_j + c) is separable through the piecewise-linear
// nonlinearity: the "negative" set per row i is a prefix of j's sorted by t_j.
// With prefix sums P0/P1 of G (and t*G) in t-sorted order:
//   (W@G)[i,:] = (s_i+c)*(SG - 0.9*P0[k_i]) + (S1 - 0.9*P1[k_i])
// where k_i = #{j : t_j < -(s_i+c)}.  This reduces the problem to O(N*F)
// memory traffic (all L2-resident: arrays are 8MB, L2 is 192MB) plus two
// 8192x256x256 GEMMs (~1.07 GFLOP each) which run on the WMMA pipes with
// V_WMMA_F32_16X16X4_F32 at full f32 fidelity.
// ---------------------------------------------------------------------------

typedef __attribute__((ext_vector_type(2))) float v2f;
typedef __attribute__((ext_vector_type(8))) float v8f;

#define NROW 8192
#define FDIM 256
#define NEG_SLOPE 0.1f
#define NCHUNK 128          // 8192 / 64
#define CHUNK 64

__device__ __forceinline__ float leaky(float z) {
    return z >= 0.0f ? z : NEG_SLOPE * z;
}

// ---------------------------------------------------------------------------
// K1: s = x@Wa, t = x@Wb, diag = leaky(s+t+c). One wave per row (wave32).
// ---------------------------------------------------------------------------
__global__ void k_st(const float* __restrict__ x, const float* __restrict__ W3,
                     const float* __restrict__ b3,
                     float* __restrict__ s, float* __restrict__ t,
                     float* __restrict__ diag) {
    int lane = threadIdx.x & 31;
    int wave = threadIdx.x >> 5;
    int row  = blockIdx.x * 8 + wave;           // grid = 1024 blocks
    const float* xr = x + row * FDIM;
    const float* Wa = W3;
    const float* Wb = W3 + FDIM;
    float a = 0.0f, b = 0.0f;
    #pragma unroll
    for (int k = lane; k < FDIM; k += 32) {
        float xv = xr[k];
        a += xv * Wa[k];
        b += xv * Wb[k];
    }
    #pragma unroll
    for (int off = 16; off > 0; off >>= 1) {
        a += __shfl_xor(a, off, 32);
        b += __shfl_xor(b, off, 32);
    }
    if (lane == 0) {
        s[row] = a;
        t[row] = b;
        diag[row] = leaky(a + b + b3[0]);
    }
}

// ---------------------------------------------------------------------------
// K2: stable rank of each t_i (O(N^2) counting sort, 67M compares -> trivial).
// perm[rank] = i, ts[rank] = t_i  (ascending, ties by index).
// ---------------------------------------------------------------------------
__global__ void k_rank(const float* __restrict__ t, float* __restrict__ ts,
                       int* __restrict__ perm) {
    __shared__ float sh[256];
    int i = blockIdx.x * 256 + threadIdx.x;     // grid = 32 blocks
    float ti = t[i];
    int rank = 0;
    for (int j0 = 0; j0 < NROW; j0 += 256) {
        __syncthreads();
        sh[threadIdx.x] = t[j0 + threadIdx.x];
        __syncthreads();
        #pragma unroll 8
        for (int j = 0; j < 256; ++j) {
            float tj = sh[j];
            rank += (tj < ti) || (tj == ti && (j0 + j) < i);
        }
    }
    perm[rank] = i;
    ts[rank] = ti;
}

// ---------------------------------------------------------------------------
// K3/K7: C = leaky(A @ B^T + bias) [+ addsrc], A:(8192,256), B:(256,256).
// WMMA f32 16x16x4. Both operands are K-major => per-lane fragments are
// contiguous float2:  frag = Op[(tile+lane&15)*256 + kb + 2*(lane>>4)].
// Each wave computes a 16(M) x 64(N) strip (4 accumulators).
// grid = 256 blocks x 256 threads (8 waves) -> 2048 waves = 512 mt x 4 ng.
// ---------------------------------------------------------------------------
__global__ void k_gemm_nt(const float* __restrict__ A, const float* __restrict__ Bm,
                          const float* __restrict__ bias,
                          const float* __restrict__ addsrc,
                          float* __restrict__ C) {
    int lane = threadIdx.x & 31;
    int wave = threadIdx.x >> 5;
    int wg   = blockIdx.x * 8 + wave;
    int mt   = wg >> 2;                 // 0..511
    int ng   = wg & 3;                  // 0..3  (64-col group)
    int m    = lane & 15;
    int half = lane >> 4;               // k-sub-pair select

    const float* arow = A + (mt * 16 + m) * FDIM + 2 * half;
    const float* brow[4];
    #pragma unroll
    for (int sub = 0; sub < 4; ++sub)
        brow[sub] = Bm + (ng * 64 + sub * 16 + m) * FDIM + 2 * half;

    v8f acc[4] = {};
    #pragma unroll 4
    for (int kb = 0; kb < FDIM; kb += 4) {
        v2f a = *(const v2f*)(arow + kb);
        #pragma unroll
        for (int sub = 0; sub < 4; ++sub) {
            v2f b = *(const v2f*)(brow[sub] + kb);
            acc[sub] = __builtin_amdgcn_wmma_f32_16x16x4_f32(
                false, a, false, b, (short)0, acc[sub], false, false);
        }
    }
    // D layout: VGPR v holds row (v + 8*half), col = lane&15 within sub-tile.
    #pragma unroll
    for (int sub = 0; sub < 4; ++sub) {
        int col = ng * 64 + sub * 16 + m;
        float bv = bias[col];
        #pragma unroll
        for (int v = 0; v < 8; ++v) {
            int orow = mt * 16 + v + 8 * half;
            float val = leaky(acc[sub][v] + bv);
            if (addsrc) val += addsrc[orow * FDIM + col];
            C[orow * FDIM + col] = val;
        }
    }
}

// ---------------------------------------------------------------------------
// K4a: per-chunk sums of G (and ts*G) in sorted order. grid = 128 x 256.
// ---------------------------------------------------------------------------
__global__ void k_chunksum(const float* __restrict__ G, const int* __restrict__ perm,
                           const float* __restrict__ ts,
                           float* __restrict__ C0, float* __restrict__ C1) {
    int cb = blockIdx.x, col = threadIdx.x;
    float s0 = 0.0f, s1 = 0.0f;
    int base = cb * CHUNK;
    #pragma unroll 4
    for (int r = 0; r < CHUNK; ++r) {
        int rank = base + r;
        float v = G[perm[rank] * FDIM + col];
        s0 += v;
        s1 += ts[rank] * v;
    }
    C0[cb * FDIM + col] = s0;
    C1[cb * FDIM + col] = s1;
}

// K4b: exclusive scan over the 128 chunk totals per column; also zero row 0
// of P0/P1. grid = 1 x 256.
__global__ void k_chunkscan(float* __restrict__ C0, float* __restrict__ C1,
                            float* __restrict__ P0, float* __restrict__ P1) {
    int col = threadIdx.x;
    float r0 = 0.0f, r1 = 0.0f;
    for (int cb = 0; cb < NCHUNK; ++cb) {
        float t0 = C0[cb * FDIM + col];
        float t1 = C1[cb * FDIM + col];
        C0[cb * FDIM + col] = r0;
        C1[cb * FDIM + col] = r1;
        r0 += t0;
        r1 += t1;
    }
    P0[col] = 0.0f;
    P1[col] = 0.0f;
}

// K4c: write full exclusive prefixes P0/P1[(rank+1)*F + col]. grid = 128 x 256.
__global__ void k_prefix(const float* __restrict__ G, const int* __restrict__ perm,
                         const float* __restrict__ ts,
                         const float* __restrict__ C0, const float* __restrict__ C1,
                         float* __restrict__ P0, float* __restrict__ P1) {
    int cb = blockIdx.x, col = threadIdx.x;
    float r0 = C0[cb * FDIM + col];
    float r1 = C1[cb * FDIM + col];
    int base = cb * CHUNK;
    for (int r = 0; r < CHUNK; ++r) {
        int rank = base + r;
        float v = G[perm[rank] * FDIM + col];
        r0 += v;
        r1 += ts[rank] * v;
        P0[(rank + 1) * FDIM + col] = r0;
        P1[(rank + 1) * FDIM + col] = r1;
    }
}

// ---------------------------------------------------------------------------
// K5: k_i = lower_bound(ts, -(s_i + c)).  grid = 32 x 256.
// ---------------------------------------------------------------------------
__global__ void k_kidx(const float* __restrict__ s, const float* __restrict__ b3,
                       const float* __restrict__ ts, int* __restrict__ kidx) {
    int i = blockIdx.x * 256 + threadIdx.x;
    float theta = -(s[i] + b3[0]);
    int lo = 0, hi = NROW;
    while (lo < hi) {
        int mid = (lo + hi) >> 1;
        if (ts[mid] < theta) lo = mid + 1; else hi = mid;
    }
    kidx[i] = lo;
}

// ---------------------------------------------------------------------------
// K6: M[i,:] = ((s_i+c)*(SG-0.9*P0[k]) + (S1-0.9*P1[k]) - diag_i*G[i,:])/8191
// grid = 8192 x 256 (pure bandwidth, ~32MB, L2-resident).
// ---------------------------------------------------------------------------
__global__ void k_assembleM(const float* __restrict__ G, const float* __restrict__ P0,
                            const float* __restrict__ P1, const float* __restrict__ s,
                            const float* __restrict__ diag, const int* __restrict__ kidx,
                            const float* __restrict__ b3, float* __restrict__ M) {
    int i = blockIdx.x, col = threadIdx.x;
    float si = s[i] + b3[0];
    float di = diag[i];
    int   k  = kidx[i];
    float SG = P0[NROW * FDIM + col];
    float S1 = P1[NROW * FDIM + col];
    float p0 = P0[k * FDIM + col];
    float p1 = P1[k * FDIM + col];
    float v = si * (SG - 0.9f * p0) + (S1 - 0.9f * p1) - di * G[i * FDIM + col];
    M[i * FDIM + col] = v * (1.0f / (float)(NROW - 1));
}

// ---------------------------------------------------------------------------
extern "C" void kernel_launch(void* const* d_in, const int* in_sizes, int n_in,
                              void* d_out, int out_size, void* d_ws, size_t ws_size,
                              hipStream_t stream) {
    (void)in_sizes; (void)n_in; (void)out_size; (void)ws_size;
    const float* x  = (const float*)d_in[0];
    const float* W3 = (const float*)d_in[1];
    const float* b3 = (const float*)d_in[2];
    const float* W6 = (const float*)d_in[3];
    const float* b6 = (const float*)d_in[4];
    const float* W5 = (const float*)d_in[5];
    const float* b5 = (const float*)d_in[6];
    float* out = (float*)d_out;

    // Workspace carve-up (~32.4 MB total required).
    char* ws = (char*)d_ws;
    size_t off = 0;
    auto alloc = [&](size_t bytes) { char* p = ws + off; off += (bytes + 255) & ~(size_t)255; return p; };
    float* s    = (float*)alloc(NROW * sizeof(float));
    float* t    = (float*)alloc(NROW * sizeof(float));
    float* diag = (float*)alloc(NROW * sizeof(float));
    float* ts   = (float*)alloc(NROW * sizeof(float));
    int*   perm = (int*)  alloc(NROW * sizeof(int));
    int*   kidx = (int*)  alloc(NROW * sizeof(int));
    float* C0   = (float*)alloc(NCHUNK * FDIM * sizeof(float));
    float* C1   = (float*)alloc(NCHUNK * FDIM * sizeof(float));
    float* G    = (float*)alloc((size_t)NROW * FDIM * sizeof(float));
    float* P0   = (float*)alloc((size_t)(NROW + 1) * FDIM * sizeof(float));
    float* P1   = (float*)alloc((size_t)(NROW + 1) * FDIM * sizeof(float));
    float* M    = (float*)alloc((size_t)NROW * FDIM * sizeof(float));

    k_st      <<<NROW / 8, 256, 0, stream>>>(x, W3, b3, s, t, diag);
    k_rank    <<<NROW / 256, 256, 0, stream>>>(t, ts, perm);
    k_gemm_nt <<<256, 256, 0, stream>>>(x, W6, b6, nullptr, G);       // G = leaky(x@W6^T + b6)
    k_chunksum<<<NCHUNK, 256, 0, stream>>>(G, perm, ts, C0, C1);
    k_chunkscan<<<1, 256, 0, stream>>>(C0, C1, P0, P1);
    k_prefix  <<<NCHUNK, 256, 0, stream>>>(G, perm, ts, C0, C1, P0, P1);
    k_kidx    <<<NROW / 256, 256, 0, stream>>>(s, b3, ts, kidx);
    k_assembleM<<<NROW, 256, 0, stream>>>(G, P0, P1, s, diag, kidx, b3, M);
    k_gemm_nt <<<256, 256, 0, stream>>>(M, W5, b5, x, out);           // out = leaky(M@W5^T + b5) + x
}